// Lossfunction_26302379721078
// MI455X (gfx1250) — compile-verified
//
#include <hip/hip_runtime.h>
#include <stdint.h>

#define C_CLASSES        100
#define SLICES_PER_BLOCK 128
#define BLOCK_THREADS    128
#define TOTAL_SLICES     512000   // 10*5*10*1024
#define B_SIZE           1024
#define TILE_DWORDS      (SLICES_PER_BLOCK * C_CLASSES)   // 12800 dwords = 51200 B
#define NEG_BIG_F        -3.0e38f

typedef float        v2f  __attribute__((ext_vector_type(2)));
typedef float        v8f  __attribute__((ext_vector_type(8)));
typedef unsigned int u32x4 __attribute__((ext_vector_type(4)));
typedef int          i32x4 __attribute__((ext_vector_type(4)));
typedef int          i32x8 __attribute__((ext_vector_type(8)));

__global__ void zero_acc_kernel(float* acc) { acc[0] = 0.0f; }

__global__ __launch_bounds__(BLOCK_THREADS)
void margin_loss_kernel(const float* __restrict__ pred,
                        const int*   __restrict__ label,
                        float*       __restrict__ acc) {
  __shared__ __align__(16) float tile[TILE_DWORDS];
  const int tid = threadIdx.x;
  const int wg  = blockIdx.x;
  const float* src = pred + (size_t)wg * TILE_DWORDS;

#if __has_builtin(__builtin_amdgcn_tensor_load_to_lds) && __has_builtin(__builtin_amdgcn_s_wait_tensorcnt)
  // --- Stage tile via Tensor Data Mover: 1-D contiguous 51200B global -> LDS ---
  if (tid < 32) {                       // wave 0 issues the DMA (EXEC ignored by TDM)
    const unsigned ndw = TILE_DWORDS;   // 12800 elements of 4 bytes
    unsigned long long ga = (unsigned long long)(uintptr_t)src;
    unsigned lds_off = (unsigned)(uintptr_t)&tile[0];   // LDS byte offset (low 32 bits of LDS aperture)
    u32x4 g0;
    g0[0] = 1u;                                          // count=1, user descriptor, no gather/iterate
    g0[1] = lds_off;                                     // lds_addr
    g0[2] = (unsigned)(ga & 0xffffffffull);              // global_addr[31:0]
    g0[3] = (unsigned)((ga >> 32) & 0x1ffffffull)        // global_addr[56:32]
          | 0x80000000u;                                 // type=2 ("image") in bits 127:126
    i32x8 g1;
    g1[0] = (int)(2u << 16);                             // data_size = 4B; wg_mask=0 (not in cluster)
    g1[1] = (int)((ndw & 0xffffu) << 16);                // tensor_dim0[15:0]
    g1[2] = (int)((ndw >> 16) | (1u << 16));             // tensor_dim0[31:16] | tensor_dim1[15:0]=1
    g1[3] = (int)((ndw & 0xffffu) << 16);                // tensor_dim1 hi=0 | tile_dim0 = 12800
    g1[4] = 0;                                           // tile_dim1 = 0 (unused), tile_dim2 = 0
    g1[5] = (int)ndw;                                    // tensor_dim0_stride[31:0]
    g1[6] = 0;                                           // stride hi | tensor_dim1_stride lo
    g1[7] = 0;
    i32x4 gz = {0, 0, 0, 0};
#if defined(__clang_major__) && (__clang_major__ >= 23)
    i32x8 gz8 = {0, 0, 0, 0, 0, 0, 0, 0};
    __builtin_amdgcn_tensor_load_to_lds(g0, g1, gz, gz, gz8, 0);
#else
    __builtin_amdgcn_tensor_load_to_lds(g0, g1, gz, gz, 0);
#endif
    __builtin_amdgcn_s_wait_tensorcnt(0);
  }
  __syncthreads();
#else
  // --- Fallback: cooperative coalesced b128 copy ---
  {
    const float4* s4 = (const float4*)src;
    float4*       t4 = (float4*)tile;
    for (int i = tid; i < TILE_DWORDS / 4; i += BLOCK_THREADS) t4[i] = s4[i];
  }
  __syncthreads();
#endif

  // --- Per-thread: one slice of 100 classes; this thread OWNS its LDS row ---
  const int slice = wg * SLICES_PER_BLOCK + tid;
  const int lab   = label[slice & (B_SIZE - 1)];
  float* row = tile + tid * C_CLASSES;                   // 400B stride, 16B aligned

  // fy by direct indexed LDS read; then knock out the true class in-place.
  // LDS ops from the same wave are in-order, so later loads observe the store.
  // (Tile is re-DMA'd from global every launch, so this mutation is replay-safe.)
  const float fy = row[lab];
  row[lab] = NEG_BIG_F;

  // fnym = plain max over the (masked) 100 values: 25 x ds_load_b128 + max tree.
  const float4* r4 = (const float4*)row;
  float4 m4 = r4[0];
#pragma unroll
  for (int i = 1; i < C_CLASSES / 4; ++i) {
    float4 v = r4[i];
    m4.x = fmaxf(m4.x, v.x);
    m4.y = fmaxf(m4.y, v.y);
    m4.z = fmaxf(m4.z, v.z);
    m4.w = fmaxf(m4.w, v.w);
  }
  const float fnym = fmaxf(fmaxf(m4.x, m4.y), fmaxf(m4.z, m4.w));

  // l = max(m + T - fy, 0) + max(m + fnym, 0), m=0.1, T=0.5
  float loss = fmaxf(0.6f - fy, 0.0f) + fmaxf(0.1f + fnym, 0.0f);

  // --- Wave32 reduction via V_WMMA_F32_16X16X4_F32 with B = ones ---
  // A(m,k): lane L<16 holds A(L,0)=loss; lane L>=16 holds A(L-16,2)=loss; other K slots 0.
  // D[m][n] = loss[m] + loss[m+16]; lane's 8-row sum + shfl_xor(16) = wave total.
#if __has_builtin(__builtin_amdgcn_wmma_f32_16x16x4_f32)
  v2f a;  a[0] = loss;  a[1] = 0.0f;
  v2f bm; bm[0] = 1.0f; bm[1] = 1.0f;
  v8f cm = {};
  v8f d = __builtin_amdgcn_wmma_f32_16x16x4_f32(false, a, false, bm, (short)0, cm, false, false);
  float s = d[0] + d[1] + d[2] + d[3] + d[4] + d[5] + d[6] + d[7];
  s += __shfl_xor(s, 16, 32);
#else
  float s = loss;
  for (int off = 16; off > 0; off >>= 1) s += __shfl_xor(s, off, 32);
#endif
  if ((tid & 31) == 0) atomicAdd(acc, s);
}

__global__ void finalize_kernel(const float* __restrict__ acc, float* __restrict__ out) {
  out[0] = acc[0] * (1.0f / (float)TOTAL_SLICES);
}

extern "C" void kernel_launch(void* const* d_in, const int* in_sizes, int n_in,
                              void* d_out, int out_size, void* d_ws, size_t ws_size,
                              hipStream_t stream) {
  const float* pred  = (const float*)d_in[0];
  const int*   label = (const int*)d_in[1];
  float* acc = (float*)d_ws;                 // 4 bytes of scratch as global accumulator
  zero_acc_kernel<<<1, 1, 0, stream>>>(acc); // deterministic re-init every call
  margin_loss_kernel<<<TOTAL_SLICES / SLICES_PER_BLOCK, BLOCK_THREADS, 0, stream>>>(pred, label, acc);
  finalize_kernel<<<1, 1, 0, stream>>>(acc, (float*)d_out);
}